// GraphConvInstanceGlobalMaxSmallSoftMaxAggr_10874857193952
// MI455X (gfx1250) — compile-verified
//
#include <hip/hip_runtime.h>
#include <math.h>

#define NN      50000      // nodes
#define NE      800000     // edges
#define NG      128        // graphs
#define INC     31         // input channels (padded to 32 on device)
#define HID     64
#define EPS     1e-5f

typedef __attribute__((ext_vector_type(2))) float v2f;
typedef __attribute__((ext_vector_type(8))) float v8f;

// ---- order-preserving f32 <-> u32 encoding for atomic max ----
static __device__ __forceinline__ unsigned enc_f32(float x) {
    unsigned b = __float_as_uint(x);
    return (b & 0x80000000u) ? ~b : (b | 0x80000000u);
}
static __device__ __forceinline__ float dec_f32(unsigned e) {
    unsigned b = (e & 0x80000000u) ? (e & 0x7FFFFFFFu) : ~e;
    return __uint_as_float(b);
}
#define ENC_NEG_INF 0x007FFFFFu   // enc(-inf)

// ---------------- pad rows with zeros: [rows,Csrc] -> [rows,Cdst] ----------------
__global__ void k_pad_rows(const float* __restrict__ s, float* __restrict__ d,
                           int rows, int Csrc, int Cdst) {
    int i = blockIdx.x * blockDim.x + threadIdx.x;
    if (i >= rows * Cdst) return;
    int r = i / Cdst, c = i % Cdst;
    d[i] = (c < Csrc) ? s[r * Csrc + c] : 0.f;
}

// ---------------- init kernels ----------------
__global__ void k_init_aggr(unsigned* __restrict__ em, float* __restrict__ num,
                            float* __restrict__ den, int n) {
    int i = blockIdx.x * blockDim.x + threadIdx.x;
    if (i < n) { em[i] = ENC_NEG_INF; num[i] = 0.f; den[i] = 0.f; }
}
__global__ void k_init_enc(unsigned* __restrict__ p, int n) {
    int i = blockIdx.x * blockDim.x + threadIdx.x;
    if (i < n) p[i] = ENC_NEG_INF;
}

// ------- edge pass 1: segment max of logits (4 channels / thread, float4 loads) -------
__global__ void k_edge_max4(const int* __restrict__ src, const int* __restrict__ dst,
                            const float* __restrict__ x, const float* __restrict__ tptr,
                            unsigned* __restrict__ em, int E, int cqshift) {
    int g = blockIdx.x * blockDim.x + threadIdx.x;
    int e = g >> cqshift;
    if (e >= E) return;
    int q  = g & ((1 << cqshift) - 1);
    int Cp = 4 << cqshift;
    float t = tptr[0];
    const float4 m = ((const float4*)(x + (size_t)src[e] * Cp))[q];
    unsigned* p = em + (size_t)dst[e] * Cp + q * 4;
    atomicMax(p + 0, enc_f32(m.x * t));
    atomicMax(p + 1, enc_f32(m.y * t));
    atomicMax(p + 2, enc_f32(m.z * t));
    atomicMax(p + 3, enc_f32(m.w * t));
}

// ------- edge pass 2: e = exp(t*m - mx); num += m*e; den += e -------
__global__ void k_edge_expsum4(const int* __restrict__ src, const int* __restrict__ dst,
                               const float* __restrict__ x, const float* __restrict__ tptr,
                               const unsigned* __restrict__ em,
                               float* __restrict__ num, float* __restrict__ den,
                               int E, int cqshift) {
    int g = blockIdx.x * blockDim.x + threadIdx.x;
    int e = g >> cqshift;
    if (e >= E) return;
    int q  = g & ((1 << cqshift) - 1);
    int Cp = 4 << cqshift;
    float t = tptr[0];
    int   d = dst[e];
    const float4 m = ((const float4*)(x + (size_t)src[e] * Cp))[q];
    size_t base = (size_t)d * Cp + q * 4;
    const unsigned* pe = em + base;
    float* pn = num + base;
    float* pd = den + base;
    float e0 = __expf(m.x * t - dec_f32(pe[0]));
    float e1 = __expf(m.y * t - dec_f32(pe[1]));
    float e2 = __expf(m.z * t - dec_f32(pe[2]));
    float e3 = __expf(m.w * t - dec_f32(pe[3]));
    atomicAdd(pn + 0, m.x * e0);  atomicAdd(pd + 0, e0);
    atomicAdd(pn + 1, m.y * e1);  atomicAdd(pd + 1, e1);
    atomicAdd(pn + 2, m.z * e2);  atomicAdd(pd + 2, e2);
    atomicAdd(pn + 3, m.w * e3);  atomicAdd(pd + 3, e3);
}

// ---------------- aggr = num / den (0 for empty segments / pad channels) ----------------
__global__ void k_aggr_div(const float* __restrict__ num, const float* __restrict__ den,
                           float* __restrict__ aggr, int n) {
    int i = blockIdx.x * blockDim.x + threadIdx.x;
    if (i >= n) return;
    float dd = den[i];
    aggr[i] = dd > 0.f ? num[i] / dd : 0.f;
}

// ---------------- WMMA GEMM: out = aggr @ Wr^T + x @ Ws^T + b ----------------
// One wave per 16x16 output tile, f32 WMMA (16x16x4), fully unrolled over K=CP.
// A layout (16x4 f32): lanes 0-15 -> (M=lane, K=k+0/k+1); lanes 16-31 -> (M=lane-16, K=k+2/k+3).
// B layout mirrors A with lane = N column. No branches before WMMA -> EXEC stays all-ones.
template<int CP>
__global__ void k_gemm(const float* __restrict__ aggr, const float* __restrict__ xin,
                       const float* __restrict__ Wr, const float* __restrict__ Ws,
                       const float* __restrict__ bias, float* __restrict__ out, int Nrows) {
    const int lane = threadIdx.x;           // 0..31
    const int half = lane >> 4;
    const int l16  = lane & 15;
    const int m0   = blockIdx.x * 16;
    const int h0   = threadIdx.y * 16;      // 4 n-tiles cover HID=64
    int arow = m0 + l16;
    if (arow >= Nrows) arow = Nrows - 1;    // clamp (select, not branch); masked on store
    const int bcol = h0 + l16;

    const float* ap = aggr + (size_t)arow * CP + half * 2;
    const float* xp = xin  + (size_t)arow * CP + half * 2;
    const float* wr = Wr   + (size_t)bcol * CP + half * 2;
    const float* wsp= Ws   + (size_t)bcol * CP + half * 2;

    v8f acc = {};
#pragma unroll
    for (int k = 0; k < CP; k += 4) {
        v2f a = *(const v2f*)(ap + k);
        v2f b = *(const v2f*)(wr + k);
        acc = __builtin_amdgcn_wmma_f32_16x16x4_f32(false, a, false, b,
                                                    (short)0, acc, false, false);
    }
#pragma unroll
    for (int k = 0; k < CP; k += 4) {
        v2f a = *(const v2f*)(xp + k);
        v2f b = *(const v2f*)(wsp + k);
        acc = __builtin_amdgcn_wmma_f32_16x16x4_f32(false, a, false, b,
                                                    (short)0, acc, false, false);
    }

    float bv = bias[bcol];
#pragma unroll
    for (int r = 0; r < 8; ++r) {
        int row = m0 + r + half * 8;        // VGPR r -> row m0+r / m0+r+8
        if (row < Nrows) out[(size_t)row * HID + bcol] = acc[r] + bv;
    }
}

// ---------------- instance norm over feature dim + ReLU, in place (one wave / row) ----------------
__global__ void k_instnorm_relu(float* __restrict__ x, int Nrows) {
    int wave = (blockIdx.x * blockDim.x + threadIdx.x) >> 5;
    int lane = threadIdx.x & 31;
    if (wave >= Nrows) return;
    float v0 = x[wave * HID + lane];
    float v1 = x[wave * HID + lane + 32];
    float s = v0 + v1;
    for (int off = 16; off; off >>= 1) s += __shfl_xor(s, off);
    float mu = s * (1.f / HID);
    float d0 = v0 - mu, d1 = v1 - mu;
    float q = d0 * d0 + d1 * d1;
    for (int off = 16; off; off >>= 1) q += __shfl_xor(q, off);
    float inv = rsqrtf(q * (1.f / HID) + EPS);
    x[wave * HID + lane]      = fmaxf(d0 * inv, 0.f);
    x[wave * HID + lane + 32] = fmaxf(d1 * inv, 0.f);
}

// ---------------- per-graph channel max pooling ----------------
__global__ void k_graph_max(const float* __restrict__ xl, const int* __restrict__ batch,
                            unsigned* __restrict__ he, int n) {
    int i = blockIdx.x * blockDim.x + threadIdx.x;
    if (i >= n) return;
    int node = i >> 6;
    int c    = i & 63;
    atomicMax(&he[batch[node] * HID + c], enc_f32(xl[i]));
}

__global__ void k_decode_h(const unsigned* __restrict__ henc, float* __restrict__ h, int n) {
    int i = blockIdx.x * blockDim.x + threadIdx.x;
    if (i >= n) return;
    int g = i / (3 * HID);
    int c = i % (3 * HID);
    int layer = c / HID, cc = c % HID;
    h[i] = dec_f32(henc[layer * NG * HID + g * HID + cc]);
}

// ---------------- tiny MLP head ----------------
__global__ void k_mlp1(const float* __restrict__ h, const float* __restrict__ Wl1,
                       const float* __restrict__ bl1, float* __restrict__ z1) {
    int i = blockIdx.x * blockDim.x + threadIdx.x;
    if (i >= NG * 2 * HID) return;
    int g = i / (2 * HID), oc = i % (2 * HID);
    float s = bl1[oc];
    for (int k = 0; k < 3 * HID; ++k) s += h[g * 3 * HID + k] * Wl1[oc * 3 * HID + k];
    z1[i] = fmaxf(s, 0.f);
}

__global__ void k_mlp2_norm(const float* __restrict__ z1, const float* __restrict__ Wl2,
                            const float* __restrict__ bl2, float* __restrict__ outp) {
    int wave = (blockIdx.x * blockDim.x + threadIdx.x) >> 5;  // one wave per graph
    int lane = threadIdx.x & 31;                              // one lane per output (32)
    if (wave >= NG) return;
    float s = bl2[lane];
    for (int k = 0; k < 2 * HID; ++k) s += z1[wave * 2 * HID + k] * Wl2[lane * 2 * HID + k];
    float q = s * s;
    for (int off = 16; off; off >>= 1) q += __shfl_xor(q, off);
    float nrm = fmaxf(sqrtf(q), 1e-12f);
    outp[wave * (HID / 2) + lane] = s / nrm;
}

// ---------------- host launch ----------------
extern "C" void kernel_launch(void* const* d_in, const int* in_sizes, int n_in,
                              void* d_out, int out_size, void* d_ws, size_t ws_size,
                              hipStream_t stream) {
    const float* x    = (const float*)d_in[0];
    const int*  eidx  = (const int*)  d_in[1];
    const int*  batch = (const int*)  d_in[2];
    const float* t    = (const float*)d_in[3];
    const float* W1r  = (const float*)d_in[4];
    const float* b1   = (const float*)d_in[5];
    const float* W1s  = (const float*)d_in[6];
    const float* W2r  = (const float*)d_in[7];
    const float* b2   = (const float*)d_in[8];
    const float* W2s  = (const float*)d_in[9];
    const float* W3r  = (const float*)d_in[10];
    const float* b3   = (const float*)d_in[11];
    const float* W3s  = (const float*)d_in[12];
    const float* Wl1  = (const float*)d_in[13];
    const float* bl1  = (const float*)d_in[14];
    const float* Wl2  = (const float*)d_in[15];
    const float* bl2  = (const float*)d_in[16];
    const int* src = eidx;
    const int* dst = eidx + NE;

    float* ws = (float*)d_ws;
    size_t o = 0;
    float*    x1     = ws + o; o += (size_t)NN * HID;
    float*    x2     = ws + o; o += (size_t)NN * HID;
    float*    x3     = ws + o; o += (size_t)NN * HID;
    unsigned* encmax = (unsigned*)(ws + o); o += (size_t)NN * HID;
    float*    num    = ws + o; o += (size_t)NN * HID;
    float*    den    = ws + o; o += (size_t)NN * HID;
    float*    aggr   = ws + o; o += (size_t)NN * HID;
    float*    xpad   = ws + o; o += (size_t)NN * 32;     // layer-1 input padded 31->32
    float*    w1rp   = ws + o; o += (size_t)HID * 32;    // W1r padded
    float*    w1sp   = ws + o; o += (size_t)HID * 32;    // W1s padded
    unsigned* henc   = (unsigned*)(ws + o); o += 3 * NG * HID;
    float*    h      = ws + o; o += NG * 3 * HID;
    float*    z1     = ws + o; o += NG * 2 * HID;

    // one-time padding copies (still per-call: deterministic, inputs unmodified)
    k_pad_rows<<<(NN * 32 + 255) / 256, 256, 0, stream>>>(x, xpad, NN, INC, 32);
    k_pad_rows<<<(HID * 32 + 255) / 256, 256, 0, stream>>>(W1r, w1rp, HID, INC, 32);
    k_pad_rows<<<(HID * 32 + 255) / 256, 256, 0, stream>>>(W1s, w1sp, HID, INC, 32);
    k_init_enc<<<(3 * NG * HID + 255) / 256, 256, 0, stream>>>(henc, 3 * NG * HID);

    struct Layer { const float* xin; int Cp; const float *Wr, *b, *Ws; float* xout; };
    Layer layers[3] = {
        { xpad, 32, w1rp, b1, w1sp, x1 },
        { x1,   64, W2r,  b2, W2s,  x2 },
        { x2,   64, W3r,  b3, W3s,  x3 },
    };

    for (int l = 0; l < 3; ++l) {
        const Layer& L = layers[l];
        int nagg = NN * L.Cp;
        k_init_aggr<<<(nagg + 255) / 256, 256, 0, stream>>>(encmax, num, den, nagg);

        int cqshift = (L.Cp == 32) ? 3 : 4;          // 4-channel groups per edge
        int tot = NE << cqshift;
        int eblocks = (tot + 255) / 256;
        k_edge_max4<<<eblocks, 256, 0, stream>>>(src, dst, L.xin, t, encmax, NE, cqshift);
        k_edge_expsum4<<<eblocks, 256, 0, stream>>>(src, dst, L.xin, t, encmax, num, den,
                                                    NE, cqshift);
        k_aggr_div<<<(nagg + 255) / 256, 256, 0, stream>>>(num, den, aggr, nagg);

        dim3 gblk(32, 4);
        if (L.Cp == 32)
            k_gemm<32><<<(NN + 15) / 16, gblk, 0, stream>>>(aggr, L.xin, L.Wr, L.Ws, L.b,
                                                            L.xout, NN);
        else
            k_gemm<64><<<(NN + 15) / 16, gblk, 0, stream>>>(aggr, L.xin, L.Wr, L.Ws, L.b,
                                                            L.xout, NN);

        k_instnorm_relu<<<(NN + 7) / 8, 256, 0, stream>>>(L.xout, NN);

        int ngm = NN * HID;
        k_graph_max<<<(ngm + 255) / 256, 256, 0, stream>>>(L.xout, batch,
                                                           henc + (size_t)l * NG * HID, ngm);
    }

    k_decode_h<<<(NG * 3 * HID + 255) / 256, 256, 0, stream>>>(henc, h, NG * 3 * HID);
    k_mlp1<<<(NG * 2 * HID + 255) / 256, 256, 0, stream>>>(h, Wl1, bl1, z1);
    k_mlp2_norm<<<NG * 32 / 128, 128, 0, stream>>>(z1, Wl2, bl2, (float*)d_out);
}